// DeepFeatureLoss_29248727286281
// MI455X (gfx1250) — compile-verified
//
#include <hip/hip_runtime.h>
#include <math.h>

#define SIGMA 0.05f
#define LOG2E 1.4426950408889634f

typedef float v2f __attribute__((ext_vector_type(2)));
typedef float v8f __attribute__((ext_vector_type(8)));

// Raw v_exp_f32 (no denormal-range fixup). Flush-to-zero below 2^-126 is the
// correct behavior for softmax tail terms here.
__device__ __forceinline__ float fexp2(float x) { return __builtin_amdgcn_exp2f(x); }

// Pre-pass (one thread per point):
//  - scale points by 1/sigma, pad [B,N,3] -> [B,N,4] so the spatial Gram is a
//    single V_WMMA_F32_16X16X4_F32 per 16x16 tile
//  - precompute per-point norm pairs:
//      colnorm[i] = (|x_i|^2, |fea2_i|^2)   (consumed per j-tile, 1x b64 load)
//      rownorm[i] = (|x_i|^2, |fea1_i|^2)   (consumed once per row strip)
//  - zero d_out
__global__ void dfl_prep_kernel(const float* __restrict__ points,
                                const float* __restrict__ fea1,
                                const float* __restrict__ fea2,
                                float* __restrict__ xs4,
                                float2* __restrict__ colnorm,
                                float2* __restrict__ rownorm,
                                float* __restrict__ out,
                                int BN, int Bout) {
  int idx = blockIdx.x * blockDim.x + threadIdx.x;
  if (idx < Bout) out[idx] = 0.0f;
  if (idx < BN) {
    const float inv = 1.0f / SIGMA;
    float x0 = points[idx * 3 + 0] * inv;
    float x1 = points[idx * 3 + 1] * inv;
    float x2 = points[idx * 3 + 2] * inv;
    xs4[idx * 4 + 0] = x0;
    xs4[idx * 4 + 1] = x1;
    xs4[idx * 4 + 2] = x2;
    xs4[idx * 4 + 3] = 0.0f;
    float xn = x0 * x0 + x1 * x1 + x2 * x2;

    const float4* f1v = (const float4*)(fea1) + (size_t)idx * 8;
    const float4* f2v = (const float4*)(fea2) + (size_t)idx * 8;
    float n1 = 0.f, n2 = 0.f;
#pragma unroll
    for (int t = 0; t < 8; ++t) {
      float4 a = f1v[t];
      float4 b = f2v[t];
      n1 += a.x * a.x + a.y * a.y + a.z * a.z + a.w * a.w;
      n2 += b.x * b.x + b.y * b.y + b.z * b.z + b.w * b.w;
    }
    colnorm[idx] = make_float2(xn, n2);
    rownorm[idx] = make_float2(xn, n1);
  }
}

// One wave32 per 16-row strip. Fused fp32-WMMA Gram + online (base-2) softmax
// second moments. Spatial softmax uses the analytically known row max of 0
// (diagonal distance = 0), so only the feature softmax carries online-max state.
__global__ __launch_bounds__(256)
void dfl_main_kernel(const float* __restrict__ xs4,
                     const float* __restrict__ fea1,
                     const float* __restrict__ fea2,
                     const float2* __restrict__ colnorm,
                     const float2* __restrict__ rownorm,
                     const float* __restrict__ weights,
                     float* __restrict__ out,
                     int N, int nstripsPerB, int totalStrips) {
  const int lane  = threadIdx.x & 31;
  const int wave  = threadIdx.x >> 5;
  const int strip = blockIdx.x * (blockDim.x >> 5) + wave;
  if (strip >= totalStrips) return;          // wave-uniform: EXEC stays all-1s for WMMA
  const int b    = strip / nstripsPerB;
  const int ib   = (strip - b * nstripsPerB) * 16;
  const int half = lane >> 4;                // 0: K/comp pair {0,1}, 1: {2,3}
  const int lm   = lane & 15;

  const float2* __restrict__ x2  = (const float2*)xs4 + (size_t)b * N * 2;   // [N][2] float2
  const float2* __restrict__ f1  = (const float2*)fea1 + (size_t)b * N * 16; // [N][16] float2
  const float2* __restrict__ f2  = (const float2*)fea2 + (size_t)b * N * 16;
  const float2* __restrict__ cn2 = colnorm + (size_t)b * N;
  const float2* __restrict__ rn2 = rownorm + (size_t)b * N;

  // ---- A fragments for rows ib..ib+15 (fixed for the whole strip) ----
  const int arow = ib + lm;
  float2 axv = x2[(size_t)arow * 2 + half];
  v2f ax; ax[0] = axv.x; ax[1] = axv.y;
  v2f af[8];
#pragma unroll
  for (int t = 0; t < 8; ++t) {
    float2 v = f1[(size_t)arow * 16 + 2 * t + half];
    af[t][0] = v.x; af[t][1] = v.y;
  }
  // row norms: C layout puts slot r at row r (lanes 0-15) / r+8 (lanes 16-31)
  float nix[8], nif[8];
#pragma unroll
  for (int r = 0; r < 8; ++r) {
    float2 v = rn2[ib + r + half * 8];
    nix[r] = v.x; nif[r] = v.y;
  }

  // moment state per (row-slot, column-lane-group):
  //   spatial (max known = 0): s1 = sum e1, spp = sum e1^2
  //   feature (online max m2): s2, sqq, and cross spq (scaled by feature side only)
  float m2[8], s1[8], s2[8], spp[8], spq[8], sqq[8];
#pragma unroll
  for (int r = 0; r < 8; ++r) {
    m2[r] = -INFINITY;
    s1[r] = 0.f; s2[r] = 0.f; spp[r] = 0.f; spq[r] = 0.f; sqq[r] = 0.f;
  }

  for (int jb = 0; jb < N; jb += 16) {
    const int col = jb + lm;
    // B fragments + precomputed column norms (pure global loads, no DS)
    float2 cn  = cn2[col];                       // (.x = |x_col|^2, .y = |fea2_col|^2)
    float2 bxv = x2[(size_t)col * 2 + half];
    v2f bx; bx[0] = bxv.x; bx[1] = bxv.y;

    v8f S = {};
    S = __builtin_amdgcn_wmma_f32_16x16x4_f32(false, ax, false, bx, (short)0, S, false, false);

    v8f F = {};
#pragma unroll
    for (int t = 0; t < 8; ++t) {
      float2 v = f2[(size_t)col * 16 + 2 * t + half];
      v2f bf; bf[0] = v.x; bf[1] = v.y;
      F = __builtin_amdgcn_wmma_f32_16x16x4_f32(false, af[t], false, bf, (short)0, F, false, false);
    }

#pragma unroll
    for (int r = 0; r < 8; ++r) {
      // logits (base-2): -(|xi|^2 + |xj|^2 - 2 xi.xj) * log2(e); both <= 0
      float l1 = (2.f * S[r] - nix[r] - cn.x) * LOG2E;
      float l2 = (2.f * F[r] - nif[r] - cn.y) * LOG2E;
      float e1  = fexp2(l1);                 // spatial max is exactly 0: no rescale
      float nm2 = fmaxf(m2[r], l2);
      float sc2 = fexp2(m2[r] - nm2);
      float e2  = fexp2(l2 - nm2);
      m2[r] = nm2;
      s1[r]  += e1;
      spp[r]  = fmaf(e1, e1, spp[r]);
      s2[r]   = fmaf(s2[r], sc2, e2);
      sqq[r]  = fmaf(sqq[r], sc2 * sc2, e2 * e2);
      spq[r]  = fmaf(spq[r], sc2, e1 * e2);
    }
  }

  // merge the 16 column-lane partial states per row (xor butterfly within each 16-half)
#pragma unroll
  for (int r = 0; r < 8; ++r) {
#pragma unroll
    for (int off = 1; off < 16; off <<= 1) {
      // spatial moments: plain sums (common max = 0)
      s1[r]  += __shfl_xor(s1[r],  off, 32);
      spp[r] += __shfl_xor(spp[r], off, 32);
      // feature-max-dependent moments
      float m2o  = __shfl_xor(m2[r],  off, 32);
      float s2o  = __shfl_xor(s2[r],  off, 32);
      float sqqo = __shfl_xor(sqq[r], off, 32);
      float spqo = __shfl_xor(spq[r], off, 32);
      float nm2 = fmaxf(m2[r], m2o);
      float ea  = fexp2(m2[r] - nm2), eb = fexp2(m2o - nm2);
      m2[r]  = nm2;
      s2[r]  = s2[r]  * ea + s2o * eb;
      sqq[r] = sqq[r] * (ea * ea) + sqqo * (eb * eb);
      spq[r] = spq[r] * ea + spqo * eb;
    }
  }

  // lanes lm==0 (lane 0 -> rows ib..ib+7, lane 16 -> rows ib+8..ib+15)
  if (lm == 0) {
    float acc = 0.f;
#pragma unroll
    for (int r = 0; r < 8; ++r) {
      int row = ib + r + half * 8;
      float inv1 = 1.0f / s1[r], inv2 = 1.0f / s2[r];
      float res = spp[r] * inv1 * inv1 - 2.f * spq[r] * (inv1 * inv2) + sqq[r] * inv2 * inv2;
      acc += res * weights[(size_t)b * N + row];
    }
    atomicAdd(&out[b], acc);
  }
}

extern "C" void kernel_launch(void* const* d_in, const int* in_sizes, int n_in,
                              void* d_out, int out_size, void* d_ws, size_t ws_size,
                              hipStream_t stream) {
  const float* points  = (const float*)d_in[0];
  const float* fea1    = (const float*)d_in[1];
  const float* fea2    = (const float*)d_in[2];
  const float* weights = (const float*)d_in[3];
  float* out = (float*)d_out;

  const int B  = out_size;            // reference returns [B]
  const int BN = in_sizes[3];         // weights is [B,N]
  const int N  = BN / B;

  // workspace layout: xs4 [BN*4 floats] | colnorm [BN float2] | rownorm [BN float2]
  float*  ws      = (float*)d_ws;
  float*  xs4     = ws;
  float2* colnorm = (float2*)(ws + (size_t)4 * BN);
  float2* rownorm = (float2*)(ws + (size_t)6 * BN);

  const int threads = 256;
  const int pblocks = (BN + threads - 1) / threads;
  dfl_prep_kernel<<<pblocks, threads, 0, stream>>>(points, fea1, fea2, xs4,
                                                   colnorm, rownorm, out, BN, B);

  const int nstripsPerB = N / 16;
  const int totalStrips = B * nstripsPerB;
  const int wavesPerBlk = threads / 32;
  const int blocks      = (totalStrips + wavesPerBlk - 1) / wavesPerBlk;
  dfl_main_kernel<<<blocks, threads, 0, stream>>>(xs4, fea1, fea2, colnorm, rownorm,
                                                  weights, out, N, nstripsPerB, totalStrips);
}